// SingleTaskGCN_65687229825243
// MI455X (gfx1250) — compile-verified
//
#include <hip/hip_runtime.h>
#include <hip/hip_bf16.h>

#define D_FEAT 128
#define C_OUT  40

typedef __attribute__((ext_vector_type(2))) float v2f;
typedef __attribute__((ext_vector_type(8))) float v8f;
typedef __attribute__((ext_vector_type(4))) unsigned int v4u;
typedef __attribute__((ext_vector_type(8))) int v8i;
typedef __attribute__((ext_vector_type(4))) int v4i;
typedef __attribute__((address_space(3))) float lds_f;

// ---------------------------------------------------------------------------
// TDM: DMA a 16x128 f32 row-major tile from global into LDS (contiguous).
// Descriptor per CDNA5 ISA.
// Issue from ONE wave (EXEC ignored by TDM); caller barriers afterwards.
// ---------------------------------------------------------------------------
__device__ __forceinline__ void tdm_load_tile16x128(const float* gsrc, float* lds_dst,
                                                    int rows_remaining) {
    unsigned lds_off = (unsigned)(unsigned long long)(lds_f*)lds_dst;   // LDS byte addr
    unsigned long long ga = (unsigned long long)gsrc;
    unsigned dim1 = (unsigned)rows_remaining;       // rows available (OOB rows read 0)

    v4u g0;
    g0[0] = 1u;                                      // count=1, is_restore=0, gather=0
    g0[1] = lds_off;                                 // lds_addr (bytes)
    g0[2] = (unsigned)(ga & 0xFFFFFFFFull);          // global_addr[31:0]
    g0[3] = (unsigned)((ga >> 32) & 0x01FFFFFFull)   // global_addr[56:32]
          | (2u << 30);                              // type=2 ("image")

    v8i g1;
    g1[0] = (int)(2u << 16);                         // wg_mask=0 | data_size=2 (4B)
    g1[1] = (int)(128u << 16);                       // tensor_dim0[15:0]=128
    g1[2] = (int)((dim1 & 0xFFFFu) << 16);           // dim0[31:16]=0 | dim1[15:0]
    g1[3] = (int)((dim1 >> 16) | (128u << 16));      // dim1[31:16] | tile_dim0=128
    g1[4] = 16;                                      // tile_dim1=16 | tile_dim2=0
    g1[5] = 128;                                     // tensor_dim0_stride[31:0]=128
    g1[6] = 0;                                       // stride0[47:32] | stride1[15:0]
    g1[7] = 0;                                       // stride1[47:16]

    v4i z4 = {0, 0, 0, 0};
#if defined(__clang_major__) && (__clang_major__ >= 23)
    v8i z8 = {0, 0, 0, 0, 0, 0, 0, 0};
    __builtin_amdgcn_tensor_load_to_lds(g0, g1, z4, z4, z8, 0);
#else
    __builtin_amdgcn_tensor_load_to_lds(g0, g1, z4, z4, 0);
#endif
    __builtin_amdgcn_s_wait_tensorcnt(0);            // tile resident before barrier
}

// ---------------------------------------------------------------------------
// Degree / normalization
// ---------------------------------------------------------------------------
__global__ void k_fill1(float* p, int n) {
    int i = blockIdx.x * blockDim.x + threadIdx.x;
    if (i < n) p[i] = 1.0f;                // self-loop contributes 1 to degree
}

__global__ void k_deg(const int* __restrict__ ei, float* __restrict__ deg, int E) {
    int e = blockIdx.x * blockDim.x + threadIdx.x;
    if (e < E) unsafeAtomicAdd(&deg[ei[(size_t)E + e]], 1.0f);  // dst = ei[1][e]
}

__global__ void k_rsqrt(float* p, int n) {
    int i = blockIdx.x * blockDim.x + threadIdx.x;
    if (i < n) p[i] = rsqrtf(p[i]);        // deg >= 1 always (self loop)
}

// ---------------------------------------------------------------------------
// WMMA GEMM: H[N,128] = X[N,128] @ W[128,128] + b
// block = 256 threads (8 waves); each block: 16 rows (TDM-staged), wave: 16 cols
// ---------------------------------------------------------------------------
__global__ void k_gemm128_wmma(const float* __restrict__ X,
                               const float* __restrict__ W,
                               const float* __restrict__ bias,
                               float* __restrict__ H, int N) {
    __shared__ float Xs[16 * D_FEAT];
    const int row0 = blockIdx.x * 16;

    if (threadIdx.x < 32)                       // wave 0 drives the TDM
        tdm_load_tile16x128(X + (size_t)row0 * D_FEAT, Xs, N - row0);
    __syncthreads();

    const int wave = threadIdx.x >> 5;
    const int lane = threadIdx.x & 31;
    const int half = lane >> 4;      // 0: lanes 0-15, 1: lanes 16-31
    const int m    = lane & 15;
    const int col  = wave * 16;

    v8f acc = {};
    #pragma unroll
    for (int k = 0; k < D_FEAT; k += 4) {
        const int ka = k + 2 * half;           // per-lane K base (ISA A-frag layout)
        v2f a, b;
        a[0] = Xs[m * D_FEAT + ka];
        a[1] = Xs[m * D_FEAT + ka + 1];
        b[0] = W[(size_t)ka * D_FEAT + col + m];
        b[1] = W[(size_t)(ka + 1) * D_FEAT + col + m];
        acc = __builtin_amdgcn_wmma_f32_16x16x4_f32(false, a, false, b,
                                                    (short)0, acc, false, false);
    }

    const float bv = bias[col + m];
    #pragma unroll
    for (int r = 0; r < 8; ++r) {
        int row = row0 + r + 8 * half;         // lanes 16-31 carry M+8
        if (row < N) H[(size_t)row * D_FEAT + col + m] = acc[r] + bv;
    }
}

// ---------------------------------------------------------------------------
// agg[i] = H[i] * dinv[i]^2   (self-loop term; also zero-initializes agg)
// ---------------------------------------------------------------------------
__global__ void k_selfloop(const float* __restrict__ H, const float* __restrict__ dinv,
                           float* __restrict__ agg, int N) {
    int idx = blockIdx.x * blockDim.x + threadIdx.x;     // N*32 float4s
    if (idx >= N * 32) return;
    int row = idx >> 5;
    float w = dinv[row] * dinv[row];
    float4 v = ((const float4*)H)[idx];
    float4 o = { v.x * w, v.y * w, v.z * w, v.w * w };
    ((float4*)agg)[idx] = o;
}

// ---------------------------------------------------------------------------
// Edge scatter: agg[dst] += H[src] * dinv[src]*dinv[dst]
// one wave32 per edge, float4 per lane (coalesced 512B gather per edge)
// ---------------------------------------------------------------------------
__global__ void k_scatter(const float* __restrict__ H, const int* __restrict__ ei,
                          const float* __restrict__ dinv, float* __restrict__ agg, int E) {
    int e = blockIdx.x * 8 + (threadIdx.x >> 5);
    if (e >= E) return;
    const int lane = threadIdx.x & 31;
    const int s = ei[e];
    const int d = ei[(size_t)E + e];
    const float nrm = dinv[s] * dinv[d];
    const float4 v = *(const float4*)(H + (size_t)s * D_FEAT + lane * 4);
    float* dst = agg + (size_t)d * D_FEAT + lane * 4;
    unsafeAtomicAdd(dst + 0, v.x * nrm);
    unsafeAtomicAdd(dst + 1, v.y * nrm);
    unsafeAtomicAdd(dst + 2, v.z * nrm);
    unsafeAtomicAdd(dst + 3, v.w * nrm);
}

// ---------------------------------------------------------------------------
// BatchNorm statistics: per-feature sum / sumsq
// ---------------------------------------------------------------------------
__global__ void k_zero_stats(float* stats) { stats[threadIdx.x] = 0.0f; }   // 256 thr

__global__ void k_bn_stats(const float* __restrict__ A, float* __restrict__ stats, int N) {
    const int f  = threadIdx.x;                    // 128 threads
    const int r0 = blockIdx.x * 64;
    const int r1 = min(r0 + 64, N);
    float s = 0.0f, q = 0.0f;
    for (int r = r0; r < r1; ++r) {
        float v = A[(size_t)r * D_FEAT + f];
        s += v; q += v * v;
    }
    unsafeAtomicAdd(&stats[f], s);
    unsafeAtomicAdd(&stats[D_FEAT + f], q);
}

__global__ void k_bn_finalize(float* stats, const float* __restrict__ g,
                              const float* __restrict__ be, float Ninv) {
    const int f = threadIdx.x;                     // 128 threads
    float mu  = stats[f] * Ninv;
    float var = stats[D_FEAT + f] * Ninv - mu * mu;
    float sc  = g[f] * rsqrtf(var + 1e-5f);
    stats[f]          = sc;                        // reuse buffer: scale
    stats[D_FEAT + f] = be[f] - mu * sc;           //               shift
}

// ---------------------------------------------------------------------------
// Fused BN * ReLU (+ residual) + row L2-normalize. One wave32 per row.
// ---------------------------------------------------------------------------
__global__ void k_bn_relu_l2(const float* __restrict__ A, const float* __restrict__ stats,
                             const float* __restrict__ resin, float* __restrict__ Xout,
                             float* __restrict__ resout, int N, int hasRes) {
    int row = blockIdx.x * 8 + (threadIdx.x >> 5);
    if (row >= N) return;
    const int lane = threadIdx.x & 31;
    const int f = lane * 4;

    float4 v  = *(const float4*)(A + (size_t)row * D_FEAT + f);
    float4 sc = *(const float4*)(stats + f);
    float4 sh = *(const float4*)(stats + D_FEAT + f);

    float x0 = fmaxf(fmaf(v.x, sc.x, sh.x), 0.0f);
    float x1 = fmaxf(fmaf(v.y, sc.y, sh.y), 0.0f);
    float x2 = fmaxf(fmaf(v.z, sc.z, sh.z), 0.0f);
    float x3 = fmaxf(fmaf(v.w, sc.w, sh.w), 0.0f);

    if (hasRes) {
        float4 r = *(const float4*)(resin + (size_t)row * D_FEAT + f);
        x0 += r.x; x1 += r.y; x2 += r.z; x3 += r.w;
    }

    float ss = x0 * x0 + x1 * x1 + x2 * x2 + x3 * x3;
    #pragma unroll
    for (int off = 16; off > 0; off >>= 1) ss += __shfl_xor(ss, off, 32);

    float inv = 1.0f / fmaxf(sqrtf(ss), 1e-12f);
    float4 o = { x0 * inv, x1 * inv, x2 * inv, x3 * inv };
    *(float4*)(Xout + (size_t)row * D_FEAT + f)   = o;
    *(float4*)(resout + (size_t)row * D_FEAT + f) = o;
}

// ---------------------------------------------------------------------------
// Output GEMM: logits[N,40] = X[N,128] @ Wout[128,40] + bout
// block = 96 threads (3 waves: col tiles 0/16/32, tile 2 masked to cols<40)
// ---------------------------------------------------------------------------
__global__ void k_gemm_out_wmma(const float* __restrict__ X, const float* __restrict__ W,
                                const float* __restrict__ bias, float* __restrict__ out, int N) {
    __shared__ float Xs[16 * D_FEAT];
    const int row0 = blockIdx.x * 16;

    if (threadIdx.x < 32)
        tdm_load_tile16x128(X + (size_t)row0 * D_FEAT, Xs, N - row0);
    __syncthreads();

    const int wave = threadIdx.x >> 5;
    const int lane = threadIdx.x & 31;
    const int half = lane >> 4;
    const int m    = lane & 15;
    const int n    = wave * 16 + m;
    const bool ok  = (n < C_OUT);

    v8f acc = {};
    #pragma unroll
    for (int k = 0; k < D_FEAT; k += 4) {
        const int ka = k + 2 * half;
        v2f a, b;
        a[0] = Xs[m * D_FEAT + ka];
        a[1] = Xs[m * D_FEAT + ka + 1];
        b[0] = ok ? W[(size_t)ka * C_OUT + n] : 0.0f;
        b[1] = ok ? W[(size_t)(ka + 1) * C_OUT + n] : 0.0f;
        acc = __builtin_amdgcn_wmma_f32_16x16x4_f32(false, a, false, b,
                                                    (short)0, acc, false, false);
    }

    if (ok) {
        const float bv = bias[n];
        #pragma unroll
        for (int r = 0; r < 8; ++r) {
            int row = row0 + r + 8 * half;
            if (row < N) out[(size_t)row * C_OUT + n] = acc[r] + bv;
        }
    }
}

// ---------------------------------------------------------------------------
// Host launcher
// ---------------------------------------------------------------------------
extern "C" void kernel_launch(void* const* d_in, const int* in_sizes, int n_in,
                              void* d_out, int out_size, void* d_ws, size_t ws_size,
                              hipStream_t stream) {
    const float* x   = (const float*)d_in[0];
    const int*   ei  = (const int*)d_in[1];
    const float* Wl[3] = { (const float*)d_in[2], (const float*)d_in[4], (const float*)d_in[6] };
    const float* bl[3] = { (const float*)d_in[3], (const float*)d_in[5], (const float*)d_in[7] };
    const float* gl[3] = { (const float*)d_in[8], (const float*)d_in[10], (const float*)d_in[12] };
    const float* el[3] = { (const float*)d_in[9], (const float*)d_in[11], (const float*)d_in[13] };
    const float* Wout = (const float*)d_in[14];
    const float* bout = (const float*)d_in[15];
    float* out = (float*)d_out;

    const int N = in_sizes[0] / D_FEAT;
    const int E = in_sizes[1] / 2;
    const size_t ND = (size_t)N * D_FEAT;

    // workspace carve (floats, 256-float aligned)
    float* ws    = (float*)d_ws;
    size_t off   = 0;
    float* dinv  = ws + off; off += ((size_t)N + 255) & ~(size_t)255;
    float* stats = ws + off; off += 256;
    float* H     = ws + off; off += ND;
    float* agg   = ws + off; off += ND;
    float* Xc    = ws + off; off += ND;
    float* res   = ws + off; off += ND;
    (void)ws_size; (void)n_in; (void)out_size;

    const int T = 256;
    // --- degree / symmetric norm ---
    k_fill1 <<<(N + T - 1) / T, T, 0, stream>>>(dinv, N);
    k_deg   <<<(E + T - 1) / T, T, 0, stream>>>(ei, dinv, E);
    k_rsqrt <<<(N + T - 1) / T, T, 0, stream>>>(dinv, N);

    const int gemmBlocks = (N + 15) / 16;
    const float Ninv = 1.0f / (float)N;
    const float* Xin = x;

    for (int l = 0; l < 3; ++l) {
        k_gemm128_wmma<<<gemmBlocks, 256, 0, stream>>>(Xin, Wl[l], bl[l], H, N);
        k_selfloop    <<<((size_t)N * 32 + T - 1) / T, T, 0, stream>>>(H, dinv, agg, N);
        k_scatter     <<<(E + 7) / 8, 256, 0, stream>>>(H, ei, dinv, agg, E);
        k_zero_stats  <<<1, 256, 0, stream>>>(stats);
        k_bn_stats    <<<(N + 63) / 64, 128, 0, stream>>>(agg, stats, N);
        k_bn_finalize <<<1, 128, 0, stream>>>(stats, gl[l], el[l], Ninv);
        k_bn_relu_l2  <<<(N + 7) / 8, 256, 0, stream>>>(agg, stats, res, Xc, res, N, l > 0);
        Xin = Xc;
    }

    k_gemm_out_wmma<<<gemmBlocks, 96, 0, stream>>>(Xc, Wout, bout, out, N);
}